// GraphAttention_9620726743550
// MI455X (gfx1250) — compile-verified
//
#include <hip/hip_runtime.h>
#include <math.h>

// Problem constants (from reference)
#define Qn   4
#define Nn   50000
#define Kn   8
#define FIN  128
#define FOUT 64
#define CH   512                      // softmax/context chunk (nodes per block)
#define PB   ((Nn + CH - 1) / CH)     // 98 partial blocks per query

typedef __attribute__((ext_vector_type(2))) float v2f;
typedef __attribute__((ext_vector_type(8))) float v8f;

#define AGG_LD 132   // padded row stride for agg tile (bank-conflict free A reads)

// ---------------------------------------------------------------------------
// Kernel 1: h[q][n][f] = tanh( mean_k q[edges[n][k]] @ W1 + b1 )
// 256 threads = 8 waves; 32 nodes per block; WMMA f32 16x16x4, K-loop of 32.
// ---------------------------------------------------------------------------
__launch_bounds__(256)
__global__ void k_conv1(const float* __restrict__ qin, const int* __restrict__ edges,
                        const float* __restrict__ W1, const float* __restrict__ b1,
                        float* __restrict__ h) {
    __shared__ float sW1[FIN * FOUT];        // 32 KB, row-major [k][n]
    __shared__ float sAgg[32 * AGG_LD];      // ~16.9 KB gather-mean tile
    __shared__ float sB1[FOUT];
    __shared__ int   sEdge[32 * Kn];

    const int qi    = blockIdx.y;
    const int nbase = blockIdx.x * 32;
    const int t     = threadIdx.x;
    const float* qbase = qin + (size_t)qi * Nn * FIN;

    // stage W1 (coalesced float4), bias, edge tile
    {
        const float4* W1v  = (const float4*)W1;
        float4*       sW1v = (float4*)sW1;
        #pragma unroll
        for (int i = t; i < (FIN * FOUT) / 4; i += 256) sW1v[i] = W1v[i];
    }
    if (t < FOUT) sB1[t] = b1[t];
    if (t < 32 * Kn) {
        int node = nbase + (t >> 3);
        sEdge[t] = (node < Nn) ? edges[node * Kn + (t & 7)] : 0;
    }
    __syncthreads();

    // cooperative gather + mean: agg[32][128]  (reads are L2-resident)
    for (int e = t; e < 32 * FIN; e += 256) {
        int node = e >> 7;           // /FIN
        int f    = e & (FIN - 1);
        const int* ep = sEdge + node * Kn;
        float s = 0.f;
        #pragma unroll
        for (int k = 0; k < Kn; ++k)
            s += qbase[(size_t)ep[k] * FIN + f];
        sAgg[node * AGG_LD + f] = s * 0.125f;
    }
    __syncthreads();

    // WMMA phase: wave w -> node-group (w>>2), feature-tile (w&3)
    const int w    = t >> 5;
    const int lane = t & 31;
    const int half = lane >> 4;          // 0: K lo pair, 1: K hi pair
    const int r    = lane & 15;
    const int ng   = w >> 2;
    const int ft   = w & 3;

    const float* aRow = sAgg + (ng * 16 + r) * AGG_LD;   // A: row M=r per lane
    const int    ncol = ft * 16 + r;                     // B/C/D column per lane

    // C layout: lane, vgpr v <-> (M = v + 8*half, N = ncol); bias is per-column
    float bias = sB1[ncol];
    v8f acc;
    #pragma unroll
    for (int i = 0; i < 8; ++i) acc[i] = bias;

    #pragma unroll
    for (int k0 = 0; k0 < FIN; k0 += 4) {
        v2f a, b;
        // A 16x4: VGPR0 = K0(lanes0-15)/K2(lanes16-31); VGPR1 = K1/K3
        a.x = aRow[k0 + 2 * half];
        a.y = aRow[k0 + 2 * half + 1];
        // B 4x16 mirror: row K striped across lanes of each VGPR
        b.x = sW1[(k0 + 2 * half) * FOUT + ncol];
        b.y = sW1[(k0 + 2 * half + 1) * FOUT + ncol];
        acc = __builtin_amdgcn_wmma_f32_16x16x4_f32(
                  false, a, false, b, (short)0, acc, false, false);
    }

    float* hq = h + (size_t)qi * Nn * FOUT;
    #pragma unroll
    for (int v = 0; v < 8; ++v) {
        int node = nbase + ng * 16 + v + 8 * half;
        if (node < Nn)
            hq[(size_t)node * FOUT + ncol] = tanhf(acc[v]);   // coalesced 16-wide rows
    }
}

// ---------------------------------------------------------------------------
// Kernel 2: score[q][n] = mean_k  h[q][edges[n][k]] . W2  + b2   (raw score out)
// ---------------------------------------------------------------------------
__launch_bounds__(256)
__global__ void k_conv2(const float* __restrict__ h, const int* __restrict__ edges,
                        const float* __restrict__ W2, const float* __restrict__ b2,
                        float* __restrict__ score) {
    __shared__ float sW2[FOUT];
    const int t = threadIdx.x;
    if (t < FOUT) sW2[t] = W2[t];
    __syncthreads();

    const int qi   = blockIdx.y;
    const int node = blockIdx.x * 256 + t;
    if (node >= Nn) return;

    const float* hq = h + (size_t)qi * Nn * FOUT;
    float s = 0.f;
    #pragma unroll
    for (int k = 0; k < Kn; ++k) {
        int idx = edges[node * Kn + k];
        const float4* hp = (const float4*)(hq + (size_t)idx * FOUT);
        #pragma unroll
        for (int f4 = 0; f4 < FOUT / 4; ++f4) {
            float4 hv = hp[f4];
            s += hv.x * sW2[f4 * 4 + 0] + hv.y * sW2[f4 * 4 + 1]
               + hv.z * sW2[f4 * 4 + 2] + hv.w * sW2[f4 * 4 + 3];
        }
    }
    score[(size_t)qi * Nn + node] = s * 0.125f + b2[0];
}

// ---------------------------------------------------------------------------
// Kernel 3a: per-(q, chunk) partial max of score
// ---------------------------------------------------------------------------
__launch_bounds__(256)
__global__ void k_pmax(const float* __restrict__ score, float* __restrict__ pmax) {
    __shared__ float red[256];
    const int qi = blockIdx.y, blk = blockIdx.x, t = threadIdx.x;
    const int base = blk * CH;
    float m = -INFINITY;
    for (int i = t; i < CH; i += 256) {
        int n = base + i;
        if (n < Nn) m = fmaxf(m, score[(size_t)qi * Nn + n]);
    }
    red[t] = m;
    __syncthreads();
    for (int s = 128; s > 0; s >>= 1) {
        if (t < s) red[t] = fmaxf(red[t], red[t + s]);
        __syncthreads();
    }
    if (t == 0) pmax[qi * PB + blk] = red[0];
}

// Kernel 3b: reduce partial maxes -> gmax[q]
__global__ void k_gmax(const float* __restrict__ pmax, float* __restrict__ gmax) {
    const int t = threadIdx.x;
    if (t < Qn) {
        float m = -INFINITY;
        for (int i = 0; i < PB; ++i) m = fmaxf(m, pmax[t * PB + i]);
        gmax[t] = m;
    }
}

// ---------------------------------------------------------------------------
// Kernel 3c: per-(q, chunk): w = exp(score-gmax); psum = sum w;
//            ctxp[f] = sum_n w[n] * values[q][n][f]   (128 threads = features)
// ---------------------------------------------------------------------------
__launch_bounds__(128)
__global__ void k_ctxpart(const float* __restrict__ score, const float* __restrict__ values,
                          const float* __restrict__ gmax, float* __restrict__ psum,
                          float* __restrict__ ctxp) {
    __shared__ float wgt[CH];
    __shared__ float red[128];
    const int qi = blockIdx.y, blk = blockIdx.x, t = threadIdx.x;
    const int base = blk * CH;
    const int cnt  = min(CH, Nn - base);
    const float gm = gmax[qi];

    float lsum = 0.f;
    for (int i = t; i < cnt; i += 128) {
        float wv = expf(score[(size_t)qi * Nn + base + i] - gm);
        wgt[i] = wv;
        lsum  += wv;
    }
    red[t] = lsum;
    __syncthreads();
    for (int s = 64; s > 0; s >>= 1) {
        if (t < s) red[t] += red[t + s];
        __syncthreads();
    }
    if (t == 0) psum[qi * PB + blk] = red[0];

    // feature-parallel accumulation; values reads coalesced across 128 lanes
    const float* vq = values + ((size_t)qi * Nn + base) * FIN;
    float acc = 0.f;
    for (int n = 0; n < cnt; ++n)
        acc += wgt[n] * vq[(size_t)n * FIN + t];
    ctxp[((size_t)qi * PB + blk) * FIN + t] = acc;
}

// Kernel 3d: context[q][f] = (sum_blk ctxp) / (sum_blk psum)
__launch_bounds__(128)
__global__ void k_finish(const float* __restrict__ psum, const float* __restrict__ ctxp,
                         float* __restrict__ ctx) {
    __shared__ float S;
    const int qi = blockIdx.x, t = threadIdx.x;
    if (t == 0) {
        float s = 0.f;
        for (int i = 0; i < PB; ++i) s += psum[qi * PB + i];
        S = s;
    }
    __syncthreads();
    float acc = 0.f;
    for (int i = 0; i < PB; ++i) acc += ctxp[((size_t)qi * PB + i) * FIN + t];
    ctx[qi * FIN + t] = acc / S;
}

// ---------------------------------------------------------------------------
// Launch. Inputs: query, values, edges, W1, b1, W2, b2.
// d_out = context[4][128] ‖ score[4][50000].
// Workspace: h (Q*N*64 f32 = 51.2 MB) + ~0.3 MB partials.
// ---------------------------------------------------------------------------
extern "C" void kernel_launch(void* const* d_in, const int* in_sizes, int n_in,
                              void* d_out, int out_size, void* d_ws, size_t ws_size,
                              hipStream_t stream) {
    (void)in_sizes; (void)n_in; (void)out_size; (void)ws_size;

    const float* query  = (const float*)d_in[0];
    const float* values = (const float*)d_in[1];
    const int*   edges  = (const int*)  d_in[2];
    const float* W1     = (const float*)d_in[3];
    const float* b1     = (const float*)d_in[4];
    const float* W2     = (const float*)d_in[5];
    const float* b2     = (const float*)d_in[6];

    float* out   = (float*)d_out;
    float* ctx   = out;                 // [Qn][FIN]
    float* score = out + Qn * FIN;      // [Qn][Nn]

    float* ws   = (float*)d_ws;
    float* h    = ws;                                   // Qn*Nn*FOUT
    float* pmax = ws + (size_t)Qn * Nn * FOUT;          // Qn*128 slots
    float* gmax = pmax + Qn * 128;                      // Qn
    float* psum = gmax + 32;                            // Qn*128 slots
    float* ctxp = psum + Qn * 128;                      // Qn*PB*FIN

    dim3 g1((Nn + 31) / 32, Qn);
    k_conv1<<<g1, 256, 0, stream>>>(query, edges, W1, b1, h);

    dim3 g2((Nn + 255) / 256, Qn);
    k_conv2<<<g2, 256, 0, stream>>>(h, edges, W2, b2, score);

    dim3 g3(PB, Qn);
    k_pmax<<<g3, 256, 0, stream>>>(score, pmax);
    k_gmax<<<1, 32, 0, stream>>>(pmax, gmax);
    k_ctxpart<<<g3, 128, 0, stream>>>(score, values, gmax, psum, ctxp);
    k_finish<<<Qn, 128, 0, stream>>>(psum, ctxp, ctx);
}